// GroupPretrainHead_85229331022143
// MI455X (gfx1250) — compile-verified
//
#include <hip/hip_runtime.h>

typedef __attribute__((ext_vector_type(2))) float v2f;
typedef __attribute__((ext_vector_type(4))) float v4f;
typedef __attribute__((ext_vector_type(8))) float v8f;

#define BATCH  8192
#define NG     8
#define DM     2048
#define MAXGS  64
#define BM     64           // samples per block (4 WMMA M-subtiles)
#define BK     32           // K-chunk staged in LDS
#define STR    (BK + 4)     // 36: gcd(36,64)=4 -> fragment reads hit all 64 banks once
#define NCHUNK (DM / BK)

__global__ void gph_zero_counts(int* counts) {
    if (threadIdx.x < NG) counts[threadIdx.x] = 0;
}

__global__ void gph_bucket(const int* __restrict__ chosen,
                           int* __restrict__ counts,
                           int* __restrict__ lists) {
    int b = blockIdx.x * blockDim.x + threadIdx.x;
    if (b < BATCH) {
        int g = chosen[b];
        int pos = atomicAdd(&counts[g], 1);
        lists[g * BATCH + pos] = b;
    }
}

__global__ __launch_bounds__(128)
void gph_gemm(const float* __restrict__ hidden,   // [B, G, D]
              const float* __restrict__ W,        // [G, 64, D] (zero-padded rows)
              const float* __restrict__ bias,     // [G, 64]    (zero-padded)
              const int*   __restrict__ gs,       // [G]
              const int*   __restrict__ counts,   // [G]
              const int*   __restrict__ lists,    // [G, B] compacted sample indices
              float* __restrict__ preds,          // [B, 64]
              float* __restrict__ valid,          // [B, 64]
              int wv) {
    __shared__ float lh[2][BM * STR];      // 2 x 64 x 36 floats = 18432 B
    __shared__ float lw[2][MAXGS * STR];   // 2 x 64 x 36 floats = 18432 B
    __shared__ int   ssidx[BM];

    const int g      = blockIdx.y;
    const int cnt    = counts[g];
    const int m_base = blockIdx.x * BM;
    if (m_base >= cnt) return;

    const int t    = threadIdx.x;
    const int lane = t & 31;
    const int wave = t >> 5;               // wave owns N-subtile [wave*16, wave*16+16)
    const int l15  = lane & 15;
    const int mhalf = (lane >> 4) * 2;     // lanes 16-31 hold K+2,K+3 of each fragment

    if (t < BM) {
        int src = m_base + t;
        if (src >= cnt) src = cnt - 1;     // duplicate last valid sample for padding rows
        ssidx[t] = lists[g * BATCH + src];
    }
    __syncthreads();

    // staging assignment: thread t stages row (t>>1), half (t&1) -> 16 floats per tile
    const int row  = t >> 1;
    const int coff = (t & 1) * (BK / 2);   // 0 or 16
    const float* hbase = hidden + ((size_t)ssidx[row] * NG + g) * DM + coff;
    const float* wbase = W + ((size_t)g * MAXGS + row) * DM + coff;

    // ---- prefetch chunk 0 and prime buffer 0 ----
    v4f ph[4], pw[4];
    #pragma unroll
    for (int j = 0; j < 4; ++j) {
        ph[j] = __builtin_nontemporal_load((const v4f*)(hbase + j * 4));
        pw[j] = *(const v4f*)(wbase + j * 4);
    }
    {
        float* lph = &lh[0][row * STR + coff];
        float* lpw = &lw[0][row * STR + coff];
        #pragma unroll
        for (int j = 0; j < 4; ++j) {
            *(v4f*)(lph + j * 4) = ph[j];
            *(v4f*)(lpw + j * 4) = pw[j];
        }
    }
    __syncthreads();

    v8f acc[4];
    #pragma unroll
    for (int i = 0; i < 4; ++i) acc[i] = (v8f){0.f,0.f,0.f,0.f,0.f,0.f,0.f,0.f};

    for (int ci = 0; ci < NCHUNK; ++ci) {
        const int cur  = ci & 1;
        const int more = (ci + 1 < NCHUNK);

        // ---- issue global loads for chunk ci+1 (overlap with WMMAs below) ----
        v4f nh[4], nw[4];
        if (more) {
            const float* hp = hbase + (ci + 1) * BK;
            const float* wp = wbase + (ci + 1) * BK;
            #pragma unroll
            for (int j = 0; j < 4; ++j) {
                nh[j] = __builtin_nontemporal_load((const v4f*)(hp + j * 4));
                nw[j] = *(const v4f*)(wp + j * 4);
            }
        }

        // ---- WMMA on LDS buffer `cur`: 4 M-subtiles x 8 k-steps ----
        #pragma unroll
        for (int mt = 0; mt < 4; ++mt) {
            const float* ha = &lh[cur][(mt * 16 + l15) * STR + mhalf];    // A: M=row, K pair
            const float* wb = &lw[cur][(wave * 16 + l15) * STR + mhalf];  // B[k][n] = W[n][k]
            v8f c = acc[mt];
            #pragma unroll
            for (int kk = 0; kk < BK; kk += 4) {
                v2f a = *(const v2f*)(ha + kk);
                v2f b = *(const v2f*)(wb + kk);
                c = __builtin_amdgcn_wmma_f32_16x16x4_f32(
                        /*neg_a=*/false, a, /*neg_b=*/false, b,
                        /*c_mod=*/(short)0, c, /*reuse_a=*/false, /*reuse_b=*/false);
            }
            acc[mt] = c;
        }

        // ---- drain prefetch into the other buffer; single barrier per iter ----
        if (more) {
            float* lph = &lh[cur ^ 1][row * STR + coff];
            float* lpw = &lw[cur ^ 1][row * STR + coff];
            #pragma unroll
            for (int j = 0; j < 4; ++j) {
                *(v4f*)(lph + j * 4) = nh[j];
                *(v4f*)(lpw + j * 4) = nw[j];
            }
        }
        __syncthreads();
    }

    // ---- epilogue: bias, mask, scatter via compacted indices ----
    const int   n     = wave * 16 + l15;
    const float bv    = bias[g * MAXGS + n];
    const int   gsz   = gs[g];
    const float vmask = (n < gsz) ? 1.0f : 0.0f;
    const int   mh    = (lane >> 4) * 8;   // C/D rows: lanes 0-15 -> M=r, lanes 16-31 -> M=r+8

    #pragma unroll
    for (int mt = 0; mt < 4; ++mt) {
        #pragma unroll
        for (int r = 0; r < 8; ++r) {
            int ml = mt * 16 + mh + r;
            if (m_base + ml < cnt) {
                int sb = ssidx[ml];
                preds[(size_t)sb * MAXGS + n] = (acc[mt][r] + bv) * vmask;
                if (wv) valid[(size_t)sb * MAXGS + n] = vmask;
            }
        }
    }
}

extern "C" void kernel_launch(void* const* d_in, const int* in_sizes, int n_in,
                              void* d_out, int out_size, void* d_ws, size_t ws_size,
                              hipStream_t stream) {
    const float* hidden = (const float*)d_in[0];   // [8192, 8, 2048] f32
    const int*   chosen = (const int*)d_in[1];     // [8192] i32
    const float* W      = (const float*)d_in[2];   // [8, 64, 2048] f32
    const float* bias   = (const float*)d_in[3];   // [8, 64] f32
    const int*   gs     = (const int*)d_in[4];     // [8] i32

    int* counts = (int*)d_ws;                      // 8 (padded to 16) ints
    int* lists  = counts + 16;                     // [8, 8192] ints = 256 KB

    float* preds = (float*)d_out;
    float* valid = preds + (size_t)BATCH * MAXGS;
    int wv = (out_size >= 2 * BATCH * MAXGS) ? 1 : 0;

    gph_zero_counts<<<1, 32, 0, stream>>>(counts);
    gph_bucket<<<(BATCH + 255) / 256, 256, 0, stream>>>(chosen, counts, lists);

    dim3 grid(BATCH / BM, NG);  // blocks beyond a group's bucket count early-exit
    gph_gemm<<<grid, 128, 0, stream>>>(hidden, W, bias, gs, counts, lists, preds, valid, wv);
}